// MultiHashtable_53635551592663
// MI455X (gfx1250) — compile-verified
//
#include <hip/hip_runtime.h>
#include <math.h>

// Multiresolution hashgrid forward (Instant-NGP style) for MI455X / gfx1250.
// Memory-gather bound: 67 MB table lives in the 192 MB L2; per (point,level)
// task we issue 8 random 8B gathers. Uses the CDNA5 async-to-LDS gather path
// (GLOBAL_LOAD_ASYNC_TO_LDS_B64 + s_wait_asynccnt) so the gathers stream into
// LDS while the wave computes the trilinear weights.

#define HG_LEVELS   16
#define HG_ENTRIES  524309u        // prime table size from reference
#define HG_BLOCK    256

// 64-bit payload type for the async b64 builtin (signature discovered from
// hipcc diagnostic: param0 = int __vector(2) * in global AS, non-const).
typedef int hg_v2i __attribute__((ext_vector_type(2)));
typedef __attribute__((address_space(1))) hg_v2i* hg_gptr;
typedef __attribute__((address_space(3))) hg_v2i* hg_lptr;

struct LevelMeta { int n; int hashed; float es; int pad; };

// Recompute per-level metadata on device each launch (deterministic, graph-safe).
// Replicates numpy: vn = int((16*1.38^i)^3); vs = (8/vn)^(1/3);
// n = int(2/vs); es = float32(2/(n-1)); hashed = vn > N_ENTRIES.
__global__ void hg_init_meta(LevelMeta* __restrict__ m) {
  int lv = threadIdx.x;
  if (lv >= HG_LEVELS) return;
  double r   = 16.0 * pow(1.38, (double)lv);
  double vnd = r * r * r;
  long long vn = (long long)vnd;                 // int() truncation
  double vs  = pow(8.0 / (double)vn, 1.0 / 3.0);
  int n      = (int)(2.0 / vs);                  // astype(int64) truncation
  LevelMeta mm;
  mm.n      = n;
  mm.hashed = (vn > (long long)HG_ENTRIES) ? 1 : 0;
  mm.es     = (float)(2.0 / (double)(n - 1));    // float64 -> float32 cast
  mm.pad    = 0;
  m[lv] = mm;
}

__global__ __launch_bounds__(HG_BLOCK) void hg_forward(
    const float* __restrict__ xyz,      // (N,3) f32 in [-1,1]
    const float* __restrict__ data,     // (L, 524309, 2) f32
    const LevelMeta* __restrict__ meta, // (L)
    float2* __restrict__ out,           // (N, L) float2  == (N, L*2) f32
    int n_points)
{
  const int tid = blockIdx.x * HG_BLOCK + threadIdx.x;
  if (tid >= n_points * HG_LEVELS) return;
  const int pt = tid >> 4;            // 16 consecutive lanes share a point
  const int lv = tid & (HG_LEVELS - 1);

  const LevelMeta mt = meta[lv];      // 16B vector load, L1-broadcast

  const float px = xyz[pt * 3 + 0];
  const float py = xyz[pt * 3 + 1];
  const float pz = xyz[pt * 3 + 2];

  // float_xyz = (xyz - lo) / es   (f32 division, as in reference)
  const float fx = (px + 1.0f) / mt.es;
  const float fy = (py + 1.0f) / mt.es;
  const float fz = (pz + 1.0f) / mt.es;

  // corner ints via float-add-then-truncate, matching (f + off).astype(int32)
  int ix0 = (int)fx,          iy0 = (int)fy,          iz0 = (int)fz;
  int ix1 = (int)(fx + 1.0f), iy1 = (int)(fy + 1.0f), iz1 = (int)(fz + 1.0f);

  // offset from UNCLIPPED corner 0 (reference order)
  const float ox = fx - (float)ix0;
  const float oy = fy - (float)iy0;
  const float oz = fz - (float)iz0;

  const int nm1 = mt.n - 1;
  ix0 = min(max(ix0, 0), nm1); iy0 = min(max(iy0, 0), nm1); iz0 = min(max(iz0, 0), nm1);
  ix1 = min(max(ix1, 0), nm1); iy1 = min(max(iy1, 0), nm1); iz1 = min(max(iz1, 0), nm1);

  // corner order matches OFFSETS: bit2 = x, bit1 = y, bit0 = z
  unsigned idx[8];
  if (mt.hashed) {
    const unsigned long long hx0 = (unsigned long long)(unsigned)ix0;
    const unsigned long long hx1 = (unsigned long long)(unsigned)ix1;
    const unsigned long long hy0 = (unsigned long long)(unsigned)iy0 * 19349663ull;
    const unsigned long long hy1 = (unsigned long long)(unsigned)iy1 * 19349663ull;
    const unsigned long long hz0 = (unsigned long long)(unsigned)iz0 * 83492791ull;
    const unsigned long long hz1 = (unsigned long long)(unsigned)iz1 * 83492791ull;
#pragma unroll
    for (int c = 0; c < 8; ++c) {
      unsigned long long h = ((c & 4) ? hx1 : hx0)
                           ^ ((c & 2) ? hy1 : hy0)
                           ^ ((c & 1) ? hz1 : hz0);
      idx[c] = (unsigned)(h % (unsigned long long)HG_ENTRIES); // const-div -> mulhi
    }
  } else {
    const unsigned n  = (unsigned)mt.n;                // dense levels: n <= 80
    const unsigned x0 = (unsigned)ix0 * n * n, x1 = (unsigned)ix1 * n * n;
    const unsigned y0 = (unsigned)iy0 * n,     y1 = (unsigned)iy1 * n;
#pragma unroll
    for (int c = 0; c < 8; ++c)
      idx[c] = ((c & 4) ? x1 : x0) + ((c & 2) ? y1 : y0)
             + ((c & 1) ? (unsigned)iz1 : (unsigned)iz0);
  }

  const float2* __restrict__ tbl = (const float2*)data + (size_t)lv * HG_ENTRIES;

  float2 v[8];
#if defined(__gfx1250__) && __has_builtin(__builtin_amdgcn_global_load_async_to_lds_b64)
  // CDNA5 async gather: stream the 8 corner float2s into LDS via the async
  // engine (ASYNCcnt), layout [corner][thread] so readback ds_load_b64 has a
  // per-lane 8B stride -> one bank pair per lane, conflict-free.
  __shared__ float2 s_val[8 * HG_BLOCK];
#pragma unroll
  for (int c = 0; c < 8; ++c) {
    __builtin_amdgcn_global_load_async_to_lds_b64(
        (hg_gptr)(tbl + idx[c]),
        (hg_lptr)&s_val[c * HG_BLOCK + threadIdx.x],
        0, 0);
  }
#endif

  // trilinear weights (computed while gathers are in flight)
  const float wx1 = fminf(fmaxf(ox, 0.0f), 1.0f);
  const float wx0 = fminf(fmaxf(1.0f - ox, 0.0f), 1.0f);
  const float wy1 = fminf(fmaxf(oy, 0.0f), 1.0f);
  const float wy0 = fminf(fmaxf(1.0f - oy, 0.0f), 1.0f);
  const float wz1 = fminf(fmaxf(oz, 0.0f), 1.0f);
  const float wz0 = fminf(fmaxf(1.0f - oz, 0.0f), 1.0f);

#if defined(__gfx1250__) && __has_builtin(__builtin_amdgcn_global_load_async_to_lds_b64)
#if __has_builtin(__builtin_amdgcn_s_wait_asynccnt)
  __builtin_amdgcn_s_wait_asynccnt(0);
#else
  asm volatile("s_wait_asynccnt 0" ::: "memory");
#endif
#pragma unroll
  for (int c = 0; c < 8; ++c) v[c] = s_val[c * HG_BLOCK + threadIdx.x];
#else
#pragma unroll
  for (int c = 0; c < 8; ++c) v[c] = tbl[idx[c]];   // global_load_b64 gathers
#endif

  float o0 = 0.0f, o1 = 0.0f;
#pragma unroll
  for (int c = 0; c < 8; ++c) {
    const float w = ((c & 4) ? wx1 : wx0) * ((c & 2) ? wy1 : wy0)
                  * ((c & 1) ? wz1 : wz0);
    o0 = fmaf(w, v[c].x, o0);
    o1 = fmaf(w, v[c].y, o1);
  }

  out[(size_t)pt * HG_LEVELS + lv] = make_float2(o0, o1); // coalesced b64 store
}

extern "C" void kernel_launch(void* const* d_in, const int* in_sizes, int n_in,
                              void* d_out, int out_size, void* d_ws, size_t ws_size,
                              hipStream_t stream) {
  const float* xyz  = (const float*)d_in[0];   // (N,3) f32
  const float* data = (const float*)d_in[1];   // (16, 524309, 2) f32
  const int n_points = in_sizes[0] / 3;

  LevelMeta* meta = (LevelMeta*)d_ws;          // 16 * 16B scratch
  hg_init_meta<<<1, 32, 0, stream>>>(meta);

  const int total  = n_points * HG_LEVELS;
  const int blocks = (total + HG_BLOCK - 1) / HG_BLOCK;
  hg_forward<<<blocks, HG_BLOCK, 0, stream>>>(xyz, data, meta,
                                              (float2*)d_out, n_points);
}